// Sampler_74105365725853
// MI455X (gfx1250) — compile-verified
//
#include <hip/hip_runtime.h>
#include <hip/hip_bf16.h>

// ---------------------------------------------------------------------------
// Gumbel-max sampler, bandwidth-roofline design for MI455X (gfx1250).
//   stochastic row: argmax_i( logits_i/T - log(max(-log(1-u_i), eps)) )
//   greedy row    : argmax_i( logits_i )
// Branch is uniform per row (scalar compare), so each block computes exactly
// one argmax kind; partials are homogeneous per row and the final fold needs
// no temperature. 131 MB single-touch stream @ 23.3 TB/s => ~5.6 us floor;
// TRANS (2x v_log_f32/elem ~ 2.6 us) and VALU (~3 us) both sit under it.
// gfx1250 paths: clause'd NT global_load_b128, global_prefetch_b8, wave32
// butterfly shuffles.
// ---------------------------------------------------------------------------

#define BB 128
#define VV 128000
#define CHUNKS 25
#define CHUNK_ELEMS (VV / CHUNKS)     // 5120 elements per block
#define CHUNK_VEC (CHUNK_ELEMS / 4)   // 1280 float4 per block
#define TPB 256
#define ITERS (CHUNK_VEC / TPB)       // exactly 5 -> full unroll, no tail
#define EPSF 1e-10f

typedef float v4f __attribute__((ext_vector_type(4)));

struct Partial { float v; int i; };   // 8 B

// Full comparator (used only in O(log) reduction steps): first index wins ties.
__device__ __forceinline__ void argmax_merge(float& bv, int& bi, float v, int i) {
    if (v > bv || (v == bv && i < bi)) { bv = v; bi = i; }
}

__global__ __launch_bounds__(TPB) void sampler_partial_kernel(
    const float* __restrict__ logits,
    const float* __restrict__ temps,
    const float* __restrict__ u,
    Partial* __restrict__ partials)
{
    const int chunk = blockIdx.x;   // 0..CHUNKS-1
    const int row   = blockIdx.y;   // 0..B-1
    const int tid   = threadIdx.x;

    const long long base = (long long)row * VV + (long long)chunk * CHUNK_ELEMS;
    const v4f* __restrict__ lg = (const v4f*)(logits + base);
    const v4f* __restrict__ uu = (const v4f*)(u + base);

    const float t = temps[row];          // scalar load, uniform branch below
    const int idx0 = chunk * CHUNK_ELEMS;

    float bv = -INFINITY; int bi = 0x7fffffff;   // single argmax accumulator

    if (t >= EPSF) {
        // ---- stochastic row: gumbel keys ----
        const float invT = 1.0f / t;
#pragma unroll
        for (int it = 0; it < ITERS; ++it) {
            const int i = tid + it * TPB;
            if (it + 2 < ITERS) {   // compile-time after unroll
                __builtin_prefetch(&lg[i + 2 * TPB], 0, 0);
                __builtin_prefetch(&uu[i + 2 * TPB], 0, 0);
            }
            v4f L = __builtin_nontemporal_load(&lg[i]);
            v4f U = __builtin_nontemporal_load(&uu[i]);
#pragma unroll
            for (int c = 0; c < 4; ++c) {
                // exponential noise: -log(1-u) via hardware v_log_f32
                const float noise = fmaxf(-__logf(1.0f - U[c]), EPSF);
                const float key = __builtin_fmaf(L[c], invT, -__logf(noise));
                // per-thread indices strictly increase -> strict '>' keeps
                // the first (lowest-index) maximum automatically
                if (key > bv) { bv = key; bi = idx0 + i * 4 + c; }
            }
        }
    } else {
        // ---- greedy row: plain argmax of logits, no transcendentals ----
#pragma unroll
        for (int it = 0; it < ITERS; ++it) {
            const int i = tid + it * TPB;
            if (it + 2 < ITERS) {
                __builtin_prefetch(&lg[i + 2 * TPB], 0, 0);
            }
            v4f L = __builtin_nontemporal_load(&lg[i]);
#pragma unroll
            for (int c = 0; c < 4; ++c) {
                if (L[c] > bv) { bv = L[c]; bi = idx0 + i * 4 + c; }
            }
        }
    }

    // wave32 butterfly reduction (gfx1250 is wave32-only): one pair only
#pragma unroll
    for (int off = 16; off > 0; off >>= 1) {
        float ov = __shfl_xor(bv, off, 32);
        int   oi = __shfl_xor(bi, off, 32);
        argmax_merge(bv, bi, ov, oi);
    }

    // cross-wave reduction through LDS (8 wave32s per block)
    __shared__ Partial s[TPB / 32];
    const int wave = tid >> 5;
    const int lane = tid & 31;
    if (lane == 0) s[wave] = Partial{bv, bi};
    __syncthreads();
    if (tid == 0) {
        Partial p = s[0];
#pragma unroll
        for (int w = 1; w < TPB / 32; ++w) {
            argmax_merge(p.v, p.i, s[w].v, s[w].i);
        }
        partials[row * CHUNKS + chunk] = p;
    }
}

__global__ __launch_bounds__(BB) void sampler_final_kernel(
    const Partial* __restrict__ partials,
    int* __restrict__ out)
{
    const int row = threadIdx.x;
    if (row >= BB) return;
    Partial p = partials[row * CHUNKS];
#pragma unroll
    for (int c = 1; c < CHUNKS; ++c) {
        Partial q = partials[row * CHUNKS + c];
        argmax_merge(p.v, p.i, q.v, q.i);
    }
    out[row] = p.i;   // partials already encode stochastic-vs-greedy per row
}

extern "C" void kernel_launch(void* const* d_in, const int* in_sizes, int n_in,
                              void* d_out, int out_size, void* d_ws, size_t ws_size,
                              hipStream_t stream) {
    const float* logits = (const float*)d_in[0];  // [B,V] f32
    const float* temps  = (const float*)d_in[1];  // [B]   f32
    const float* u      = (const float*)d_in[2];  // [B,V] f32
    Partial* partials   = (Partial*)d_ws;         // B*CHUNKS*8B = 25.6 KB
    int* out            = (int*)d_out;            // [B] int32 indices

    dim3 grid(CHUNKS, BB);
    sampler_partial_kernel<<<grid, TPB, 0, stream>>>(logits, temps, u, partials);
    sampler_final_kernel<<<1, BB, 0, stream>>>(partials, out);
}